// GATEncoder_19816979103875
// MI455X (gfx1250) — compile-verified
//
#include <hip/hip_runtime.h>
#include <math.h>

// ---------------------------------------------------------------------------
// GAT encoder for MI455X (gfx1250, wave32, WMMA).
//   - dense projections: V_WMMA_F32_16X16X4_F32, 2 N-tiles per wave,
//     B panel staged in LDS via async global->LDS DMA
//   - edge softmax: atomicMax on ordered-uint encoding + global_atomic_add_f32
//   - message aggregation: float4 gathers + fp32 atomics
// ---------------------------------------------------------------------------

typedef __attribute__((ext_vector_type(2))) float v2f;
typedef __attribute__((ext_vector_type(8))) float v8f;

#define HEADS 4
#define NEG_SLOPE 0.2f

#if defined(__HIP_DEVICE_COMPILE__) &&                                         \
    __has_builtin(__builtin_amdgcn_global_load_async_to_lds_b128) &&           \
    __has_builtin(__builtin_amdgcn_s_wait_asynccnt)
#define HAVE_ASYNC_LDS 1
typedef int v4i __attribute__((vector_size(16)));
typedef __attribute__((address_space(1))) v4i* gv4i_p;   // global v4i*
typedef __attribute__((address_space(3))) v4i* lv4i_p;   // LDS v4i*
#else
#define HAVE_ASYNC_LDS 0
#endif

// ----- monotonic uint encoding of float for atomicMax-based segment max -----
__device__ __forceinline__ unsigned enc_f32(float f) {
    unsigned u = __float_as_uint(f);
    return (u & 0x80000000u) ? ~u : (u | 0x80000000u);
}
__device__ __forceinline__ float dec_f32(unsigned u) {
    return (u & 0x80000000u) ? __uint_as_float(u & 0x7FFFFFFFu)
                             : __uint_as_float(~u);
}

__device__ __forceinline__ void atomAddF(float* p, float v) {
    unsafeAtomicAdd(p, v);   // -> global_atomic_add_f32 on gfx1250
}

// ---------------------------------------------------------------------------
// WMMA f32 GEMM:  D[M,Nout] = act( A[M,K] @ B[K,Nout] + dbias + resid )
// One 16x32 output slab per wave (two 16x16 WMMA tiles sharing the A frag).
// 8 waves / block share a [K x 32] B panel in LDS.
// grid = ( ceil(ceil(M/16)/8), Nout/32 ).  K % 4 == 0, 8 <= K <= 256.
// ---------------------------------------------------------------------------
template<int HAS_DBIAS, int HAS_RESID, int RELU>
__global__ __launch_bounds__(256)
void wmma_gemm_kernel(const float* __restrict__ A, const float* __restrict__ B,
                      float* __restrict__ D, int M, int K, int Nout,
                      const float* __restrict__ dbias,   // len Nout
                      const float* __restrict__ resid)   // [M,Nout]
{
    __shared__ __align__(16) float Bs[256 * 32];   // K x 32 panel

    const int tid   = threadIdx.x;
    const int nbase = blockIdx.y * 32;

    // ---- stage B panel into LDS (16B chunks; async DMA path if available) --
    const int nchunks = K * 8;                     // float4 chunks in panel
    for (int idx = tid; idx < nchunks; idx += 256) {
        const int kk = idx >> 3, c4 = (idx & 7) << 2;
        const float* gsrc = B + (size_t)kk * Nout + nbase + c4;
        float* ldst = &Bs[kk * 32 + c4];
#if HAVE_ASYNC_LDS
        __builtin_amdgcn_global_load_async_to_lds_b128((gv4i_p)gsrc,
                                                       (lv4i_p)ldst, 0, 0);
#else
        *(float4*)ldst = *(const float4*)gsrc;
#endif
    }
#if HAVE_ASYNC_LDS
    __builtin_amdgcn_s_wait_asynccnt(0);
#endif
    __syncthreads();

    const int wid   = tid >> 5;
    const int lane  = tid & 31;
    const int half  = lane >> 4;           // 0: lanes 0-15, 1: lanes 16-31
    const int m     = lane & 15;
    const int tileM = blockIdx.x * 8 + wid;
    const int row   = tileM * 16 + m;
    // OOB rows read row 0; they only affect OOB output rows (never stored)
    const float* Arow = A + (size_t)(row < M ? row : 0) * K;

    v8f acc0 = {}, acc1 = {};
    v2f a_cur;                              // A frag: VGPR0=K{0,2}, VGPR1=K{1,3}
    a_cur.x = Arow[2 * half];
    a_cur.y = Arow[2 * half + 1];

    int k = 0;
    for (; k + 4 < K; k += 4) {
        v2f a_nxt;                          // prefetch next A frag
        a_nxt.x = Arow[k + 4 + 2 * half];
        a_nxt.y = Arow[k + 4 + 2 * half + 1];
        const int kb = (k + 2 * half) * 32 + m;
        v2f b0, b1;
        b0.x = Bs[kb];        b0.y = Bs[kb + 32];
        b1.x = Bs[kb + 16];   b1.y = Bs[kb + 48];
        acc0 = __builtin_amdgcn_wmma_f32_16x16x4_f32(
                   false, a_cur, false, b0, (short)0, acc0, false, false);
        acc1 = __builtin_amdgcn_wmma_f32_16x16x4_f32(
                   false, a_cur, false, b1, (short)0, acc1, false, false);
        a_cur = a_nxt;
    }
    {   // final k-step (no prefetch past end of row)
        const int kb = (k + 2 * half) * 32 + m;
        v2f b0, b1;
        b0.x = Bs[kb];        b0.y = Bs[kb + 32];
        b1.x = Bs[kb + 16];   b1.y = Bs[kb + 48];
        acc0 = __builtin_amdgcn_wmma_f32_16x16x4_f32(
                   false, a_cur, false, b0, (short)0, acc0, false, false);
        acc1 = __builtin_amdgcn_wmma_f32_16x16x4_f32(
                   false, a_cur, false, b1, (short)0, acc1, false, false);
    }

    const int col0 = nbase + m;
    const int col1 = nbase + 16 + m;
    const float db0 = HAS_DBIAS ? dbias[col0] : 0.0f;
    const float db1 = HAS_DBIAS ? dbias[col1] : 0.0f;
#pragma unroll
    for (int v = 0; v < 8; ++v) {
        const int rout = tileM * 16 + v + 8 * half;   // C/D layout: M = v (+8)
        if (rout < M) {
            float v0 = acc0[v] + db0;
            float v1 = acc1[v] + db1;
            if (HAS_RESID) {
                v0 += resid[(size_t)rout * Nout + col0];
                v1 += resid[(size_t)rout * Nout + col1];
            }
            if (RELU) { v0 = fmaxf(v0, 0.0f); v1 = fmaxf(v1, 0.0f); }
            D[(size_t)rout * Nout + col0] = v0;
            D[(size_t)rout * Nout + col1] = v1;
        }
    }
}

// ---------------------------------------------------------------------------
// bias2[n] = bl[n] + sum_k bb[k] * Wl[k][n]   (rank-1 fold of the input bias)
__global__ void bias_fold_kernel(const float* __restrict__ bb,
                                 const float* __restrict__ Wl,
                                 const float* __restrict__ bl,
                                 float* __restrict__ bias2, int K, int Nout)
{
    int n = blockIdx.x * blockDim.x + threadIdx.x;
    if (n >= Nout) return;
    float s = bl[n];
    for (int k = 0; k < K; ++k) s += bb[k] * Wl[(size_t)k * Nout + n];
    bias2[n] = s;
}

__global__ void fill_u32_kernel(unsigned* __restrict__ p, int count, unsigned v)
{
    int t = blockIdx.x * blockDim.x + threadIdx.x;
    if (t < count) p[t] = v;
}

// build src/dst with self-loops appended
__global__ void build_edges_kernel(const int* __restrict__ edge_index,
                                   int* __restrict__ src, int* __restrict__ dst,
                                   int E, int EP)
{
    int t = blockIdx.x * blockDim.x + threadIdx.x;
    if (t >= EP) return;
    if (t < E) { src[t] = edge_index[t]; dst[t] = edge_index[E + t]; }
    else       { src[t] = t - E;         dst[t] = t - E; }
}

// per-(node,head) attention dot products
__global__ void attn_scores_kernel(const float* __restrict__ xl,
                                   const float* __restrict__ att_s,
                                   const float* __restrict__ att_d,
                                   float* __restrict__ as_out,
                                   float* __restrict__ ad_out,
                                   int N, int Ch)
{
    int t = blockIdx.x * blockDim.x + threadIdx.x;
    if (t >= N * HEADS) return;
    int n = t / HEADS, h = t % HEADS;
    const float* xp = xl + (size_t)n * HEADS * Ch + (size_t)h * Ch;
    const float* sp = att_s + h * Ch;
    const float* dp = att_d + h * Ch;
    float ss = 0.0f, sd = 0.0f;
    for (int c = 0; c < Ch; c += 4) {
        const float4 xv = *(const float4*)(xp + c);
        const float4 sv = *(const float4*)(sp + c);
        const float4 dv = *(const float4*)(dp + c);
        ss += xv.x * sv.x + xv.y * sv.y + xv.z * sv.z + xv.w * sv.w;
        sd += xv.x * dv.x + xv.y * dv.y + xv.z * dv.z + xv.w * dv.w;
    }
    as_out[t] = ss;
    ad_out[t] = sd;
}

// pass 1: e = leaky_relu(a_src[s] + a_dst[d]); segment max via atomicMax(uint)
__global__ void edge_score_kernel(const int* __restrict__ src,
                                  const int* __restrict__ dst,
                                  const float* __restrict__ as_buf,
                                  const float* __restrict__ ad_buf,
                                  float* __restrict__ ebuf,
                                  unsigned* __restrict__ mkey, int EP)
{
    int t = blockIdx.x * blockDim.x + threadIdx.x;
    if (t >= EP) return;
    int s = src[t], d = dst[t];
#pragma unroll
    for (int h = 0; h < HEADS; ++h) {
        float v = as_buf[s * HEADS + h] + ad_buf[d * HEADS + h];
        v = v > 0.0f ? v : v * NEG_SLOPE;
        ebuf[(size_t)t * HEADS + h] = v;
        atomicMax(&mkey[d * HEADS + h], enc_f32(v));
    }
}

// pass 2: p = exp(e - max); segment sum of p
__global__ void edge_exp_kernel(const int* __restrict__ dst,
                                float* __restrict__ ebuf,
                                const unsigned* __restrict__ mkey,
                                float* __restrict__ denom, int EP)
{
    int t = blockIdx.x * blockDim.x + threadIdx.x;
    if (t >= EP) return;
    int d = dst[t];
#pragma unroll
    for (int h = 0; h < HEADS; ++h) {
        float v  = ebuf[(size_t)t * HEADS + h];
        float mm = dec_f32(mkey[d * HEADS + h]);
        float p  = __expf(v - mm);
        ebuf[(size_t)t * HEADS + h] = p;
        atomAddF(&denom[d * HEADS + h], p);
    }
}

// pass 3: gat[d] += xl[src] * alpha   (4 channels per thread, float4 gather)
__global__ void edge_agg_kernel(const int* __restrict__ src,
                                const int* __restrict__ dst,
                                const float* __restrict__ ebuf,
                                const float* __restrict__ denom,
                                const float* __restrict__ xl,
                                float* __restrict__ gat,
                                int EP, int Ch, int tpe /* = H*Ch/4 */)
{
    long t = (long)blockIdx.x * blockDim.x + threadIdx.x;
    if (t >= (long)EP * tpe) return;
    int e  = (int)(t / tpe);
    int j  = (int)(t % tpe);
    int c0 = j * 4;
    int h  = c0 / Ch;
    int s  = src[e], d = dst[e];
    int HC = HEADS * Ch;
    float alpha = ebuf[(size_t)e * HEADS + h] /
                  (denom[d * HEADS + h] + 1e-16f);
    const float4 xv = *(const float4*)(xl + (size_t)s * HC + c0);
    float* gp = gat + (size_t)d * HC + c0;
    atomAddF(gp + 0, xv.x * alpha);
    atomAddF(gp + 1, xv.y * alpha);
    atomAddF(gp + 2, xv.z * alpha);
    atomAddF(gp + 3, xv.w * alpha);
}

// ---------------------------------------------------------------------------
// host-side: one GAT layer (+ fused linear/bias/relu/residual epilogue)
// ---------------------------------------------------------------------------
static void run_layer(const float* xin, int Fin, int Ch,
                      const float* W, const float* as_w, const float* ad_w,
                      const float* bb, const float* Wl, const float* bl,
                      const float* resid, float* out,
                      float* xl, float* gat, float* as_buf, float* ad_buf,
                      unsigned* mkey, float* denom, float* ebuf, float* bias2,
                      const int* src, const int* dst,
                      int N, int EP, hipStream_t stream)
{
    const int HC = HEADS * Ch;
    const int tilesM = (N + 15) / 16;

    // xl = xin @ W
    dim3 g1((tilesM + 7) / 8, HC / 32);
    wmma_gemm_kernel<0, 0, 0><<<g1, 256, 0, stream>>>(
        xin, W, xl, N, Fin, HC, nullptr, nullptr);

    // bias2 = bb @ Wl + bl  (rank-1 fold; same stream, runs after the above)
    bias_fold_kernel<<<1, 64, 0, stream>>>(bb, Wl, bl, bias2, HC, Ch);

    int t = N * HEADS;
    attn_scores_kernel<<<(t + 255) / 256, 256, 0, stream>>>(
        xl, as_w, ad_w, as_buf, ad_buf, N, Ch);

    fill_u32_kernel<<<(N * HEADS + 255) / 256, 256, 0, stream>>>(mkey, N * HEADS, 0u);
    fill_u32_kernel<<<(N * HEADS + 255) / 256, 256, 0, stream>>>(
        (unsigned*)denom, N * HEADS, 0u);
    fill_u32_kernel<<<(N * HC + 255) / 256, 256, 0, stream>>>(
        (unsigned*)gat, N * HC, 0u);

    edge_score_kernel<<<(EP + 255) / 256, 256, 0, stream>>>(
        src, dst, as_buf, ad_buf, ebuf, mkey, EP);
    edge_exp_kernel<<<(EP + 255) / 256, 256, 0, stream>>>(
        dst, ebuf, mkey, denom, EP);

    const int tpe = HC / 4;
    long tot = (long)EP * tpe;
    edge_agg_kernel<<<(unsigned)((tot + 255) / 256), 256, 0, stream>>>(
        src, dst, ebuf, denom, xl, gat, EP, Ch, tpe);

    // out = relu( gat @ Wl + bias2 [+ resid] )
    dim3 g2((tilesM + 7) / 8, Ch / 32);
    if (resid)
        wmma_gemm_kernel<1, 1, 1><<<g2, 256, 0, stream>>>(
            gat, Wl, out, N, HC, Ch, bias2, resid);
    else
        wmma_gemm_kernel<1, 0, 1><<<g2, 256, 0, stream>>>(
            gat, Wl, out, N, HC, Ch, bias2, nullptr);
}

extern "C" void kernel_launch(void* const* d_in, const int* in_sizes, int n_in,
                              void* d_out, int out_size, void* d_ws, size_t ws_size,
                              hipStream_t stream)
{
    const int IN = 128, OUT = 64, C1 = 32;
    const int N  = in_sizes[0] / IN;
    const int E  = in_sizes[1] / 2;
    const int EP = E + N;

    const float* x   = (const float*)d_in[0];
    const int* eidx  = (const int*)d_in[1];
    const float *W0  = (const float*)d_in[2],  *as0 = (const float*)d_in[3],
                *ad0 = (const float*)d_in[4],  *b0  = (const float*)d_in[5],
                *Wl0 = (const float*)d_in[6],  *bl0 = (const float*)d_in[7],
                *W1  = (const float*)d_in[8],  *as1 = (const float*)d_in[9],
                *ad1 = (const float*)d_in[10], *b1  = (const float*)d_in[11],
                *Wl1 = (const float*)d_in[12], *bl1 = (const float*)d_in[13],
                *W2  = (const float*)d_in[14], *as2 = (const float*)d_in[15],
                *ad2 = (const float*)d_in[16], *b2  = (const float*)d_in[17],
                *Wl2 = (const float*)d_in[18], *bl2 = (const float*)d_in[19];

    // ---- workspace carve-up ----
    char* p = (char*)d_ws;
    int*      src    = (int*)p;       p += (size_t)EP * 4;
    int*      dst    = (int*)p;       p += (size_t)EP * 4;
    float*    xl     = (float*)p;     p += (size_t)N * 256 * 4;
    float*    gat    = (float*)p;     p += (size_t)N * 256 * 4;
    float*    as_buf = (float*)p;     p += (size_t)N * HEADS * 4;
    float*    ad_buf = (float*)p;     p += (size_t)N * HEADS * 4;
    unsigned* mkey   = (unsigned*)p;  p += (size_t)N * HEADS * 4;
    float*    denom  = (float*)p;     p += (size_t)N * HEADS * 4;
    float*    ebuf   = (float*)p;     p += (size_t)EP * HEADS * 4;
    float*    x0     = (float*)p;     p += (size_t)N * OUT * 4;
    float*    x1     = (float*)p;     p += (size_t)N * C1 * 4;
    float*    bias2  = (float*)p;     p += 256;
    (void)ws_size; (void)n_in; (void)out_size;

    build_edges_kernel<<<(EP + 255) / 256, 256, 0, stream>>>(eidx, src, dst, E, EP);

    // layer 0: 128 -> H x 64, then Wl0 -> 64, relu   => x0
    run_layer(x, IN, OUT, W0, as0, ad0, b0, Wl0, bl0, nullptr, x0,
              xl, gat, as_buf, ad_buf, mkey, denom, ebuf, bias2,
              src, dst, N, EP, stream);
    // layer 1: 64 -> H x 32, then Wl1 -> 32, relu    => x1
    run_layer(x0, OUT, C1, W1, as1, ad1, b1, Wl1, bl1, nullptr, x1,
              xl, gat, as_buf, ad_buf, mkey, denom, ebuf, bias2,
              src, dst, N, EP, stream);
    // layer 2: 32 -> H x 64, then Wl2 -> 64, +x0, relu => d_out
    run_layer(x1, C1, OUT, W2, as2, ad2, b2, Wl2, bl2, x0, (float*)d_out,
              xl, gat, as_buf, ad_buf, mkey, denom, ebuf, bias2,
              src, dst, N, EP, stream);
}